// CONTRASTLoss_28741921145436
// MI455X (gfx1250) — compile-verified
//
#include <hip/hip_runtime.h>
#include <math.h>

// ---------------------------------------------------------------------------
// CDNA5 (gfx1250) wave32 helpers
// ---------------------------------------------------------------------------
typedef float v2f __attribute__((ext_vector_type(2)));
typedef float v8f __attribute__((ext_vector_type(8)));

// Full 32-lane sum using the f32 WMMA pipe (exact f32 adds, no precision loss).
// A (16x4): VGPR0 = per-lane value v  -> A[m][0] = v[lane m], A[m][2] = v[lane m+16]
//           VGPR1 = 0                 -> A[m][1] = A[m][3]   = 0
// B (4x16) = all ones.
// D[m][n] = rowsum(A[m]) = v[m] + v[m+16]   (same for every n)
// D layout: lanes 0-15 hold rows 0-7 in d[0..7]; lanes 16-31 hold rows 8-15.
// Sum d[0..7] per lane, then add across wave halves -> every lane has the total.
__device__ __forceinline__ float wave_sum32(float v) {
    v2f a; a[0] = v;    a[1] = 0.0f;
    v2f b; b[0] = 1.0f; b[1] = 1.0f;
    v8f c = {};
    v8f d = __builtin_amdgcn_wmma_f32_16x16x4_f32(
        /*neg_a=*/false, a, /*neg_b=*/false, b,
        /*c_mod=*/(short)0, c, /*reuse_a=*/false, /*reuse_b=*/false);
    float s = ((d[0] + d[1]) + (d[2] + d[3])) + ((d[4] + d[5]) + (d[6] + d[7]));
    s += __shfl_xor(s, 16, 32);   // combine low/high wave halves
    return s;
}

// ---------------------------------------------------------------------------
// Kernel 0: zero the 5 f64 global accumulators (d_ws is poisoned by harness)
// ---------------------------------------------------------------------------
__global__ void contrast_init_kernel(double* __restrict__ acc) {
    int t = threadIdx.x;
    if (t < 5) acc[t] = 0.0;
}

// ---------------------------------------------------------------------------
// Kernel 1: bandwidth-bound streaming reduction.
//   per-thread f32 accumulation (small magnitudes, safe)
//   -> per-wave WMMA reduction
//   -> per-block LDS combine
//   -> global f64 atomic add (avoids cancellation error in cov later)
// ---------------------------------------------------------------------------
__global__ void __launch_bounds__(256)
contrast_reduce_kernel(const float* __restrict__ p1,
                       const float* __restrict__ p2,
                       double* __restrict__ acc, int n) {
    const int nvec   = n >> 2;               // float4 count
    const int idx    = blockIdx.x * blockDim.x + threadIdx.x;
    const int stride = gridDim.x * blockDim.x;

    const float4* __restrict__ v1 = (const float4*)p1;
    const float4* __restrict__ v2 = (const float4*)p2;

    float s1 = 0.f, s2 = 0.f, q1 = 0.f, q2 = 0.f, pp = 0.f;

    for (int i = idx; i < nvec; i += stride) {
        float4 a = v1[i];
        float4 b = v2[i];
        s1 += (a.x + a.y) + (a.z + a.w);
        s2 += (b.x + b.y) + (b.z + b.w);
        q1 = fmaf(a.x, a.x, q1); q1 = fmaf(a.y, a.y, q1);
        q1 = fmaf(a.z, a.z, q1); q1 = fmaf(a.w, a.w, q1);
        q2 = fmaf(b.x, b.x, q2); q2 = fmaf(b.y, b.y, q2);
        q2 = fmaf(b.z, b.z, q2); q2 = fmaf(b.w, b.w, q2);
        pp = fmaf(a.x, b.x, pp); pp = fmaf(a.y, b.y, pp);
        pp = fmaf(a.z, b.z, pp); pp = fmaf(a.w, b.w, pp);
    }
    // scalar tail (n % 4), converges before the WMMA below
    for (int i = (nvec << 2) + idx; i < n; i += stride) {
        float a = p1[i], b = p2[i];
        s1 += a; s2 += b;
        q1 = fmaf(a, a, q1);
        q2 = fmaf(b, b, q2);
        pp = fmaf(a, b, pp);
    }

    // Wave-level reduction on the matrix pipe (EXEC is all-1s here: no divergence).
    s1 = wave_sum32(s1);
    s2 = wave_sum32(s2);
    q1 = wave_sum32(q1);
    q2 = wave_sum32(q2);
    pp = wave_sum32(pp);

    // Block-level combine: one LDS row per wave (wave32 -> 8 waves at 256 thr).
    __shared__ float ws[8][5];
    const int wave = threadIdx.x >> 5;
    const int lane = threadIdx.x & 31;
    if (lane == 0) {
        ws[wave][0] = s1; ws[wave][1] = s2;
        ws[wave][2] = q1; ws[wave][3] = q2;
        ws[wave][4] = pp;
    }
    __syncthreads();
    if (threadIdx.x < 5) {
        const int nw = blockDim.x >> 5;
        float t = 0.f;
        for (int w = 0; w < nw; ++w) t += ws[w][threadIdx.x];
        atomicAdd(&acc[threadIdx.x], (double)t);   // global_atomic_add_f64
    }
}

// ---------------------------------------------------------------------------
// Kernel 2: finalize scalar loss in f64 (cov suffers catastrophic cancellation
// in f32: terms ~4e6, result ~0).
// ---------------------------------------------------------------------------
__global__ void contrast_finalize_kernel(const double* __restrict__ acc,
                                         float* __restrict__ out, int n) {
    double s1 = acc[0], s2 = acc[1], q1 = acc[2], q2 = acc[3], p = acc[4];
    double dn  = (double)n;
    double mu1 = s1 / dn;
    double mu2 = s2 / dn;
    double m1  = mu1 + 0.001;        // reference adds 0.001 only for the cov means
    double m2  = mu2 + 0.001;
    double inv = 1.0 / (dn - 1.0);   // ddof = 1 (torch.std unbiased)
    double var1 = (q1 - s1 * mu1) * inv;
    double var2 = (q2 - s2 * mu2) * inv;
    double std1 = sqrt(var1);
    double std2 = sqrt(var2);
    // sum((d1-m1)(d2-m2)) = p - m2*s1 - m1*s2 + n*m1*m2
    double cov  = (p - m2 * s1 - m1 * s2 + dn * m1 * m2) * inv;
    double cor  = cov / (std1 * std2 + 0.001);
    double t    = cor + 0.001;
    out[0] = (float)(0.5 * t * t);
}

// ---------------------------------------------------------------------------
// Launcher
// ---------------------------------------------------------------------------
extern "C" void kernel_launch(void* const* d_in, const int* in_sizes, int n_in,
                              void* d_out, int out_size, void* d_ws, size_t ws_size,
                              hipStream_t stream) {
    (void)n_in; (void)out_size; (void)ws_size;
    const float* d1 = (const float*)d_in[0];
    const float* d2 = (const float*)d_in[1];
    double* acc = (double*)d_ws;
    float*  out = (float*)d_out;
    const int n = in_sizes[0];

    contrast_init_kernel<<<1, 32, 0, stream>>>(acc);

    const int threads = 256;                       // 8 wave32 waves / block
    const int nvec = n >> 2;
    int blocks = (nvec + threads * 8 - 1) / (threads * 8);   // ~8 float4 iters/thread
    if (blocks < 1) blocks = 1;
    if (blocks > 8192) blocks = 8192;              // keep grid-stride loop bounded
    contrast_reduce_kernel<<<blocks, threads, 0, stream>>>(d1, d2, acc, n);

    contrast_finalize_kernel<<<1, 1, 0, stream>>>(acc, out, n);
}